// WellAwareLSTM_43782896615686
// MI455X (gfx1250) — compile-verified
//
#include <hip/hip_runtime.h>

// ---------------------------------------------------------------------------
// WellAwareLSTM on MI455X (gfx1250), f32 WMMA path (V_WMMA_F32_16X16X4_F32).
//
//   1) gemm_bias_wmma : xg0 = x @ W_ih0^T + (b_ih0+b_hh0)      [8192 x 1024]
//   2) scan_init + lstm_scan_mw (4 WGs, global barrier per step) -> y0, hn0, cn0
//   3) gemm_bias_wmma : xg1 = y0 @ W_ih1^T + (b_ih1+b_hh1)
//   4) scan_init + lstm_scan_mw -> y1 (= d_out), hn1, cn1
//
// d_ws layout: xg     [8192*1024] f32   32 MB (reused by layer 1)
//              y0     [8192* 256] f32    8 MB
//              hglob  [2][16][256] f32  32 KB (double-buffered h state in L2)
//              bar    1 x u32           (global step barrier counter)
// ---------------------------------------------------------------------------

typedef __attribute__((ext_vector_type(2))) float v2f;
typedef __attribute__((ext_vector_type(8))) float v8f;

#define L2E 1.4426950408889634f
#define NWG 4          // workgroups cooperating on the scan
#define HSTATE 4096    // 16*256 floats per h buffer

__device__ __forceinline__ float fast_sigmoid(float x) {
    float e = __builtin_amdgcn_exp2f(-x * L2E);       // overflow -> +inf -> 0, safe
    return 1.0f / (1.0f + e);
}

__device__ __forceinline__ float fast_tanh(float x) {
    float ax = __builtin_fabsf(x);
    float a  = __builtin_amdgcn_exp2f(-2.0f * L2E * ax);   // (0,1], overflow-safe
    float t  = (1.0f - a) / (1.0f + a);
    return __builtin_copysignf(t, x);
}

// out[M,N] = A[M,K] @ W[N,K]^T + (b0[N]+b1[N]).
// One 16x64 output strip per wave: 4 accumulators share each A fragment
// (5 x 8B loads per 4 WMMAs instead of 8).
__global__ __launch_bounds__(256) void gemm_bias_wmma(
    const float* __restrict__ A, const float* __restrict__ W,
    const float* __restrict__ b0, const float* __restrict__ b1,
    float* __restrict__ out, int M, int K, int N)
{
    const int lane = threadIdx.x & 31;
    const int wave = threadIdx.x >> 5;
    const int l16  = lane & 15;
    const int hw   = lane >> 4;

    const int nT   = N >> 6;                       // number of 64-col strips
    const int tile = blockIdx.x * 8 + wave;
    const int m0   = (tile / nT) << 4;
    const int n0   = (tile % nT) << 6;

    const float* arow = A + (size_t)(m0 + l16) * K + 2 * hw;
    const float* w0   = W + (size_t)(n0 + l16) * K + 2 * hw;
    const float* w1   = w0 + (size_t)16 * K;
    const float* w2   = w0 + (size_t)32 * K;
    const float* w3   = w0 + (size_t)48 * K;

    v8f a0 = {}, a1 = {}, a2 = {}, a3 = {};
    for (int k = 0; k < K; k += 8) {
#pragma unroll
        for (int kk = 0; kk < 8; kk += 4) {
            v2f a = *(const v2f*)(arow + k + kk);
            a0 = __builtin_amdgcn_wmma_f32_16x16x4_f32(false, a, false, *(const v2f*)(w0 + k + kk), (short)0, a0, false, false);
            a1 = __builtin_amdgcn_wmma_f32_16x16x4_f32(false, a, false, *(const v2f*)(w1 + k + kk), (short)0, a1, false, false);
            a2 = __builtin_amdgcn_wmma_f32_16x16x4_f32(false, a, false, *(const v2f*)(w2 + k + kk), (short)0, a2, false, false);
            a3 = __builtin_amdgcn_wmma_f32_16x16x4_f32(false, a, false, *(const v2f*)(w3 + k + kk), (short)0, a3, false, false);
        }
    }

    v8f accs[4] = {a0, a1, a2, a3};
#pragma unroll
    for (int s = 0; s < 4; ++s) {
        const int n = n0 + 16 * s + l16;
        const float bias = b0[n] + b1[n];
#pragma unroll
        for (int r = 0; r < 8; ++r)
            out[(size_t)(m0 + r + 8 * hw) * N + n] = accs[s][r] + bias;
    }
}

// Zero h0 state and the global barrier counter before each scan.
__global__ void scan_init(float* __restrict__ hglob, unsigned* __restrict__ bar) {
    const int i = blockIdx.x * blockDim.x + threadIdx.x;
    if (i < HSTATE) hglob[i] = 0.0f;
    if (i == 0) *bar = 0u;
}

// Multi-workgroup recurrence: NWG=4 blocks x 512 threads.
// Block wg owns h columns [64*wg, 64*wg+64). Wave w (0..15): gate = w>>2
// (i,f,g,o), h-tile = w&3 -> exactly one 16x16 gate tile = 64 WMMAs/step.
// Per step: stage h (L2->LDS), GEMM, LDS gate exchange, waves 0..3 do the
// cell math (c lives in their VGPRs), store h slice to L2, global barrier.
__global__ __launch_bounds__(512) void lstm_scan_mw(
    const float* __restrict__ xg,   // [B=16, T, 1024] input gates (+biases)
    const float* __restrict__ Whh,  // [1024, 256]
    float* __restrict__ y,          // [B, T, 256]
    float* __restrict__ hn, float* __restrict__ cn,   // [B, 256]
    float* __restrict__ hglob,      // [2][16][256]
    unsigned* __restrict__ bar, int T)
{
    __shared__ float hl[16][260];        // padded: stride 260 -> conflict-free b64 reads
    __shared__ float gacc[16][16][18];   // gate-tile exchange, padded stride 18

    const int tid  = threadIdx.x;
    const int lane = tid & 31;
    const int wave = tid >> 5;           // 0..15
    const int l16  = lane & 15;
    const int hw   = lane >> 4;
    const int wg   = blockIdx.x;         // 0..NWG-1

    const int gate = wave >> 2;          // 0..3 : i,f,g,o
    const int ht   = wave & 3;           // h tile inside this WG's 64-col slice
    const int ncol = gate * 256 + wg * 64 + ht * 16;   // gate-space column base

    const float* wrow = Whh + (size_t)(ncol + l16) * 256 + 2 * hw;

    v8f creg = {};                       // cell state (meaningful for waves 0..3)

    for (int t = 0; t < T; ++t) {
        const int rb = t & 1;

        // ---- stage h_prev: 4096 floats, 8 per thread, into padded LDS ----
        {
            const int i0  = tid * 8;
            const int row = i0 >> 8, col = i0 & 255;
            const float4* src = (const float4*)(hglob + rb * HSTATE + row * 256 + col);
            float4* dst = (float4*)(&hl[row][col]);
            dst[0] = src[0];
            dst[1] = src[1];
        }
        __syncthreads();

        // ---- accumulator init from precomputed input gates ----
        v8f acc;
#pragma unroll
        for (int r = 0; r < 8; ++r)
            acc[r] = xg[((size_t)(r + 8 * hw) * T + t) * 1024 + ncol + l16];

        // ---- gates += h_prev @ Whh^T over K=256 : 64 WMMAs ----
#pragma unroll 8
        for (int kk = 0; kk < 64; ++kk) {
            v2f a = *(const v2f*)(&hl[l16][kk * 4 + 2 * hw]);
            v2f b = *(const v2f*)(wrow + kk * 4);
            acc = __builtin_amdgcn_wmma_f32_16x16x4_f32(false, a, false, b, (short)0, acc, false, false);
        }

        // ---- exchange gate tiles through LDS ----
#pragma unroll
        for (int r = 0; r < 8; ++r)
            gacc[wave][r + 8 * hw][l16] = acc[r];
        __syncthreads();

        // ---- waves 0..3: LSTM cell for h tile `wave` ----
        if (wave < 4) {
            const int col = wg * 64 + wave * 16 + l16;
#pragma unroll
            for (int r = 0; r < 8; ++r) {
                const int b = r + 8 * hw;
                float ig = fast_sigmoid(gacc[ 0 + wave][b][l16]);
                float fg = fast_sigmoid(gacc[ 4 + wave][b][l16]);
                float gg = fast_tanh   (gacc[ 8 + wave][b][l16]);
                float og = fast_sigmoid(gacc[12 + wave][b][l16]);
                float c  = fg * creg[r] + ig * gg;
                creg[r]  = c;
                float h  = og * fast_tanh(c);
                hglob[(rb ^ 1) * HSTATE + b * 256 + col] = h;
                y[((size_t)b * T + t) * 256 + col] = h;
                if (t == T - 1) {        // uniform branch
                    hn[b * 256 + col] = h;
                    cn[b * 256 + col] = c;
                }
            }
            __threadfence();             // h stores device-visible before arrival
        }
        __syncthreads();

        // ---- global step barrier across the NWG workgroups ----
        if (tid == 0) {
            __hip_atomic_fetch_add(bar, 1u, __ATOMIC_RELEASE, __HIP_MEMORY_SCOPE_AGENT);
            const unsigned target = (unsigned)NWG * (unsigned)(t + 1);
            while (__hip_atomic_load(bar, __ATOMIC_ACQUIRE, __HIP_MEMORY_SCOPE_AGENT) < target)
                __builtin_amdgcn_s_sleep(1);
        }
        __syncthreads();                 // acquire + WGP$ inv done; proceed
    }
}

extern "C" void kernel_launch(void* const* d_in, const int* in_sizes, int n_in,
                              void* d_out, int out_size, void* d_ws, size_t ws_size,
                              hipStream_t stream) {
    (void)in_sizes; (void)n_in; (void)out_size; (void)ws_size;

    const float* x     = (const float*)d_in[0];
    const float* W_ih0 = (const float*)d_in[1];
    const float* W_hh0 = (const float*)d_in[2];
    const float* b_ih0 = (const float*)d_in[3];
    const float* b_hh0 = (const float*)d_in[4];
    const float* W_ih1 = (const float*)d_in[5];
    const float* W_hh1 = (const float*)d_in[6];
    const float* b_ih1 = (const float*)d_in[7];
    const float* b_hh1 = (const float*)d_in[8];

    const int B = 16, T = 512, H = 256, IN = 2560;
    const int M = B * T;                            // 8192 GEMM rows

    float* out = (float*)d_out;
    float* y1  = out;                               // [B,T,H]
    float* hn  = out + (size_t)M * H;               // h_n [2,B,H]
    float* cn  = hn + 2 * B * H;                    // c_n [2,B,H]

    float*    xg    = (float*)d_ws;                 // [8192,1024]
    float*    y0    = xg + (size_t)M * 4 * H;       // [8192,256]
    float*    hglob = y0 + (size_t)M * H;           // [2][16][256]
    unsigned* bar   = (unsigned*)(hglob + 2 * HSTATE);

    const int gblocks = (M / 16) * (4 * H / 64) / 8;     // 1024 blocks, 8 waves each

    gemm_bias_wmma<<<gblocks, 256, 0, stream>>>(x,  W_ih0, b_ih0, b_hh0, xg, M, IN, 4 * H);
    scan_init<<<16, 256, 0, stream>>>(hglob, bar);
    lstm_scan_mw<<<NWG, 512, 0, stream>>>(xg, W_hh0, y0, hn, cn, hglob, bar, T);
    gemm_bias_wmma<<<gblocks, 256, 0, stream>>>(y0, W_ih1, b_ih1, b_hh1, xg, M, H, 4 * H);
    scan_init<<<16, 256, 0, stream>>>(hglob, bar);
    lstm_scan_mw<<<NWG, 512, 0, stream>>>(xg, W_hh1, y1, hn + B * H, cn + B * H, hglob, bar, T);
}